// FourierShellCorrelationLoss_81140522156480
// MI455X (gfx1250) — compile-verified
//
#include <hip/hip_runtime.h>
#include <hip/hip_bf16.h>
#include <math.h>

// ---------------------------------------------------------------------------
// Fourier Shell Correlation loss, MI455X / gfx1250.
// 32x (1024x1024) complex FFT2 done as WMMA GEMMs: 1024 = 16 * 64
//   stage1: W16(16x16) x X(16x64)       -> V_WMMA_F32_16X16X4_F32 chains
//   twiddle: * W1024^{k1*n2}            -> LDS table + FMA
//   stage2: C(16x64) x W64(64x64)       -> V_WMMA_F32_16X16X4_F32 chains
// B-operand matrices (X, W64) live in LDS in a pair-interleaved layout so a
// WMMA B fragment {M[k][n], M[k+1][n]} is one aligned ds_load_b64 into an
// adjacent VGPR pair (no v_mov packing). Tile loops are unroll-1 so only one
// accumulator pair (16 VGPRs) is live -> no spills.
// Row pass writes a coalesced complex intermediate; column pass fuses the
// |F|^2 shell binning (LDS histogram + global f32 atomics).
// ---------------------------------------------------------------------------

typedef __attribute__((ext_vector_type(2))) float v2f;
typedef __attribute__((ext_vector_type(8))) float v8f;

#define NBATCH 16
#define NPIX   1024
#define NBINS  128
#define HW     ((size_t)NPIX * NPIX)

// pair-interleaved index for B-operand matrices with 64 columns:
// (row, col) -> ((row/2)*64 + col)*2 + row%2 ; pairs (row, row+1) contiguous.
__device__ __forceinline__ int pidx(int row, int col) {
  return (((row >> 1) * 64 + col) << 1) + (row & 1);
}

// shared-memory float layout (both FFT kernels)
#define SM_W16RE   0                    // 16x16 row-major (A operand)
#define SM_W16IM   256
#define SM_W64RE   512                  // 64x64 pair-interleaved (B operand)
#define SM_W64IM   4608
#define SM_TWRE    8704                 // W1024^{k1*n2}, 16x64 row-major
#define SM_TWIM    9728
#define SM_WAVE    10752                // 8 waves * (X 2048 + C 2048)
#define SM_BINS    (10752 + 8 * 4096)   // 129 histogram slots (pass 2)
#define SM_FLOATS  (SM_BINS + 132)
#define SMEM_BYTES (SM_FLOATS * 4)

static __device__ __forceinline__ v8f wmma4(v2f a, v2f b, v8f c) {
  // D = A * B + C ; 16x16x4 f32, wave32
  return __builtin_amdgcn_wmma_f32_16x16x4_f32(false, a, false, b, (short)0, c,
                                               false, false);
}

static __device__ __forceinline__ v2f ldsv2(const float* p) {
  return *(const v2f*)p;  // 8B-aligned LDS pair load (ds_load_b64)
}

// One wave performs one full 1024-point complex FFT.
// Xre/Xim: per-wave LDS input, PAIR-INTERLEAVED X[n1][n2] (n = 64*n1+n2).
//          Free after stage 1; caller may reuse for output staging.
// Cre/Cim: per-wave LDS stage-1 result, row-major 16x64 (A operand).
// emit(k, re, im) receives output element k = k1 + 16*k2.
template <class Emit>
static __device__ __forceinline__ void fft1024_wave(
    const float* __restrict__ Xre, const float* __restrict__ Xim,
    float* __restrict__ Cre, float* __restrict__ Cim,
    const float* __restrict__ W16re, const float* __restrict__ W16im,
    const float* __restrict__ W64re, const float* __restrict__ W64im,
    const float* __restrict__ TWre, const float* __restrict__ TWim,
    int lane, Emit emit) {
  const int m16 = lane & 15;
  const int half = lane >> 4;

  // A fragments of W16 (M=k1, K=n1), row-major contiguous pairs
  v2f a16r[4], a16i[4], a16in[4];
#pragma unroll
  for (int kc = 0; kc < 4; ++kc) {
    int k0 = 4 * kc + 2 * half;  // even
    a16r[kc] = ldsv2(W16re + m16 * 16 + k0);
    a16i[kc] = ldsv2(W16im + m16 * 16 + k0);
    a16in[kc] = -a16i[kc];  // f32 WMMA has no A-neg modifier
  }

  // ---- stage 1 (tile-at-a-time): D1 = W16 x X, twiddle, store to C ----
#pragma unroll 1
  for (int t = 0; t < 4; ++t) {
    v8f cr = {}, ci = {};
#pragma unroll
    for (int kc = 0; kc < 4; ++kc) {
      int krow = 4 * kc + 2 * half;  // even
      int off = pidx(krow, t * 16 + m16);
      v2f br = ldsv2(Xre + off);
      v2f bi = ldsv2(Xim + off);
      // complex GEMM: Dr += Ar*Br - Ai*Bi ; Di += Ar*Bi + Ai*Br
      cr = wmma4(a16r[kc], br, cr);
      cr = wmma4(a16in[kc], bi, cr);
      ci = wmma4(a16r[kc], bi, ci);
      ci = wmma4(a16i[kc], br, ci);
    }
#pragma unroll
    for (int r = 0; r < 8; ++r) {
      int m = r + 8 * half;            // k1
      int n2 = t * 16 + m16;           // n2
      int idx = m * 64 + n2;
      float twc = TWre[idx], tws = TWim[idx];
      float dr = cr[r], di = ci[r];
      Cre[idx] = dr * twc - di * tws;
      Cim[idx] = dr * tws + di * twc;
    }
  }

  // ---- stage 2: D2 = C(16x64) x W64(64x64) (M=k1, K=n2, N=k2) ----
#pragma unroll 1
  for (int t2 = 0; t2 < 4; ++t2) {
    v8f cr = {}, ci = {};
#pragma unroll
    for (int k0 = 0; k0 < 64; k0 += 4) {
      int kk = k0 + 2 * half;  // even
      v2f ar = ldsv2(Cre + m16 * 64 + kk);
      v2f ai = ldsv2(Cim + m16 * 64 + kk);
      v2f ain = -ai;
      int off = pidx(kk, t2 * 16 + m16);
      v2f br = ldsv2(W64re + off);
      v2f bi = ldsv2(W64im + off);
      cr = wmma4(ar, br, cr);
      cr = wmma4(ain, bi, cr);
      ci = wmma4(ar, bi, ci);
      ci = wmma4(ai, br, ci);
    }
#pragma unroll
    for (int r = 0; r < 8; ++r) {
      int m = r + 8 * half;        // k1
      int k2 = t2 * 16 + m16;      // k2
      emit(m + 16 * k2, cr[r], ci[r]);
    }
  }
}

static __device__ __forceinline__ void build_tables(float* smem, int tid) {
  const float w16 = -6.283185307179586f / 16.0f;
  for (int i = tid; i < 256; i += 256) {
    int m = i >> 4, n = i & 15;
    float s, c;
    __sincosf(w16 * (float)(m * n), &s, &c);
    smem[SM_W16RE + i] = c;  // row-major
    smem[SM_W16IM + i] = s;
  }
  const float w64 = -6.283185307179586f / 64.0f;
  for (int i = tid; i < 4096; i += 256) {
    int m = i >> 6, n = i & 63;
    float s, c;
    __sincosf(w64 * (float)(m * n), &s, &c);
    int p = pidx(m, n);  // pair-interleaved (B operand)
    smem[SM_W64RE + p] = c;
    smem[SM_W64IM + p] = s;
  }
  const float w1024 = -6.283185307179586f / 1024.0f;
  for (int i = tid; i < 1024; i += 256) {
    int m = i >> 6, n = i & 63;   // k1 in [0,16), n2 in [0,64)
    float s, c;
    __sincosf(w1024 * (float)(m * n), &s, &c);
    smem[SM_TWRE + i] = c;  // row-major
    smem[SM_TWIM + i] = s;
  }
}

// ---------------------------------------------------------------------------
// Pass 1: row FFTs of tc = target and dc = z - target.
// grid: 16 batches * 128 row-groups; block: 256 (8 waves, 1 row per wave).
// Intermediate images: [0..15] = tc (norm), [16..31] = dc (err).
// ---------------------------------------------------------------------------
__global__ void fsc_fft_rows(const float2* __restrict__ z,
                             const float2* __restrict__ tgt,
                             float2* __restrict__ interm) {
  extern __shared__ float smem[];
  const int tid = threadIdx.x;
  const int wave = tid >> 5;
  const int lane = tid & 31;
  build_tables(smem, tid);
  __syncthreads();

  float* Xre = smem + SM_WAVE + wave * 4096;
  float* Xim = Xre + 1024;
  float* Cre = Xre + 2048;
  float* Cim = Xre + 3072;

  const int b = blockIdx.x >> 7;
  const int row = (blockIdx.x & 127) * 8 + wave;
  const size_t base = ((size_t)b * NPIX + row) * NPIX;  // float2 elements

  // ---- tc = target ----
#pragma unroll 4
  for (int i = 0; i < 32; ++i) {
    int n = i * 32 + lane;
    float2 tv = tgt[base + n];
    int p = pidx(n >> 6, n & 63);
    Xre[p] = tv.x;
    Xim[p] = tv.y;
  }
  {
    // stage-2 output restaged into the (dead) X buffer at linear k,
    // then flushed coalesced.
    fft1024_wave(Xre, Xim, Cre, Cim, smem + SM_W16RE, smem + SM_W16IM,
                 smem + SM_W64RE, smem + SM_W64IM, smem + SM_TWRE,
                 smem + SM_TWIM, lane, [&](int k, float re, float im) {
                   Xre[k] = re;
                   Xim[k] = im;
                 });
    float2* out = interm + (size_t)b * HW + (size_t)row * NPIX;
#pragma unroll 4
    for (int i = 0; i < 32; ++i) {
      int n = i * 32 + lane;
      out[n] = make_float2(Xre[n], Xim[n]);
    }
  }

  // ---- dc = z - target ----
#pragma unroll 4
  for (int i = 0; i < 32; ++i) {
    int n = i * 32 + lane;
    float2 zv = z[base + n];
    float2 tv = tgt[base + n];
    int p = pidx(n >> 6, n & 63);
    Xre[p] = zv.x - tv.x;
    Xim[p] = zv.y - tv.y;
  }
  {
    fft1024_wave(Xre, Xim, Cre, Cim, smem + SM_W16RE, smem + SM_W16IM,
                 smem + SM_W64RE, smem + SM_W64IM, smem + SM_TWRE,
                 smem + SM_TWIM, lane, [&](int k, float re, float im) {
                   Xre[k] = re;
                   Xim[k] = im;
                 });
    float2* out = interm + (size_t)(16 + b) * HW + (size_t)row * NPIX;
#pragma unroll 4
    for (int i = 0; i < 32; ++i) {
      int n = i * 32 + lane;
      out[n] = make_float2(Xre[n], Xim[n]);
    }
  }
}

// ---------------------------------------------------------------------------
// Pass 2: column FFTs + fused |F|^2 shell binning.
// grid: 32 images * 128 col-groups; block: 256 (8 waves, 1 column per wave).
// acc layout: [arr(2)][bin(129)][batch(16)] f32, arr0=norm(tc), arr1=err(dc).
// ---------------------------------------------------------------------------
__global__ void fsc_fft_cols_bins(const float2* __restrict__ interm,
                                  float* __restrict__ acc) {
  extern __shared__ float smem[];
  const int tid = threadIdx.x;
  const int wave = tid >> 5;
  const int lane = tid & 31;
  build_tables(smem, tid);
  float* bins = smem + SM_BINS;
  for (int i = tid; i < 129; i += 256) bins[i] = 0.0f;
  __syncthreads();

  float* Xre = smem + SM_WAVE + wave * 4096;
  float* Xim = Xre + 1024;
  float* Cre = Xre + 2048;
  float* Cim = Xre + 3072;

  const int img = blockIdx.x >> 7;
  const int col = (blockIdx.x & 127) * 8 + wave;
  const float2* src = interm + (size_t)img * HW + col;

  for (int i = 0; i < 32; ++i) {
    int n = i * 32 + lane;
    float2 v = src[(size_t)n * NPIX];
    int p = pidx(n >> 6, n & 63);
    Xre[p] = v.x;
    Xim[p] = v.y;
  }

  const float ux = ((float)col - 512.0f) * (1.0f / 512.0f);
  const float ux2 = ux * ux;
  fft1024_wave(Xre, Xim, Cre, Cim, smem + SM_W16RE, smem + SM_W16IM,
               smem + SM_W64RE, smem + SM_W64IM, smem + SM_TWRE, smem + SM_TWIM,
               lane, [&](int k, float re, float im) {
                 float mag2 = re * re + im * im;
                 float uy = ((float)k - 512.0f) * (1.0f / 512.0f);
                 float e = uy * uy + ux2;
                 int bin = (int)(sqrtf(e) * 128.0f);
                 bin = bin > 128 ? 128 : bin;
                 atomicAdd(&bins[bin], mag2);  // ds_add_f32
               });
  __syncthreads();

  const int arr = img >> 4;
  const int bb = img & 15;
  for (int i = tid; i < 129; i += 256)
    atomicAdd(&acc[arr * (129 * 16) + i * 16 + bb], bins[i]);
}

// ---------------------------------------------------------------------------
// Pass 3: loss = sum_bins w[bin] * sum_b err/max(norm,1e-12) / (128*16)
// ---------------------------------------------------------------------------
__global__ void fsc_finalize(const float* __restrict__ acc,
                             const float* __restrict__ weights,
                             float* __restrict__ out) {
  __shared__ float red[256];
  const int tid = threadIdx.x;
  float partial = 0.0f;
  for (int idx = tid; idx < NBINS * NBATCH; idx += 256) {
    int bin = idx >> 4;
    int b = idx & 15;
    float nrm = acc[0 * (129 * 16) + bin * 16 + b];
    float err = acc[1 * (129 * 16) + bin * 16 + b];
    partial += weights[bin] * err / fmaxf(nrm, 1e-12f);
  }
  red[tid] = partial;
  __syncthreads();
  for (int s = 128; s > 0; s >>= 1) {
    if (tid < s) red[tid] += red[tid + s];
    __syncthreads();
  }
  if (tid == 0) out[0] = red[0] / (float)(NBINS * NBATCH);
}

extern "C" void kernel_launch(void* const* d_in, const int* in_sizes, int n_in,
                              void* d_out, int out_size, void* d_ws,
                              size_t ws_size, hipStream_t stream) {
  const float2* z = (const float2*)d_in[0];     // [16,1024,1024] complex f32
  const float2* tgt = (const float2*)d_in[1];   // [16,1024,1024] complex f32
  const float* weights = (const float*)d_in[2]; // [128] f32

  float* acc = (float*)d_ws;                               // 2*129*16 f32
  float2* interm = (float2*)((char*)d_ws + 32768);         // 32 complex images

  (void)hipFuncSetAttribute((const void*)fsc_fft_rows,
                            hipFuncAttributeMaxDynamicSharedMemorySize,
                            SMEM_BYTES);
  (void)hipFuncSetAttribute((const void*)fsc_fft_cols_bins,
                            hipFuncAttributeMaxDynamicSharedMemorySize,
                            SMEM_BYTES);

  (void)hipMemsetAsync(d_ws, 0, 2 * 129 * 16 * sizeof(float), stream);

  fsc_fft_rows<<<NBATCH * 128, 256, SMEM_BYTES, stream>>>(z, tgt, interm);
  fsc_fft_cols_bins<<<32 * 128, 256, SMEM_BYTES, stream>>>(interm, acc);
  fsc_finalize<<<1, 256, 0, stream>>>(acc, weights, (float*)d_out);
}